// HeteroGATLayer_89258010345539
// MI455X (gfx1250) — compile-verified
//
#include <hip/hip_runtime.h>
#include <hip/hip_bf16.h>

// ---------------------------------------------------------------------------
// Two-layer GATConv (PyG semantics, add_self_loops, concat heads) + broadcast
// GEMMs use V_WMMA_F32_16X16X4_F32 with 2x4 register tiling (8 WMMA / k-step).
// Aggregation is atomic-free via on-the-fly CSR (counting sort by dst).
// ---------------------------------------------------------------------------

typedef __attribute__((ext_vector_type(2))) float v2f;
typedef __attribute__((ext_vector_type(8))) float v8f;

#define NEG_SLOPE 0.2f

__device__ __forceinline__ float lrelu(float v) { return v > 0.f ? v : NEG_SLOPE * v; }

__device__ __forceinline__ void atomicMaxF(float* addr, float val) {
    unsigned int* ua = (unsigned int*)addr;
    unsigned int old = __float_as_uint(*addr);
    while (val > __uint_as_float(old)) {
        unsigned int assumed = old;
        old = atomicCAS(ua, assumed, __float_as_uint(val));
        if (old == assumed) break;
    }
}

// ------------------------------ GEMM (WMMA) --------------------------------
// C[M,N] = A[M,K] * B[K,N], row-major fp32. One wave computes a 32x64 tile
// (2 M-tiles x 4 N-tiles of 16x16) -> 8 v_wmma_f32_16x16x4_f32 per k-step.
// Requires M % 32 == 0, N % 64 == 0, K % 4 == 0.
__global__ __launch_bounds__(256) void gemm_wmma_f32(
    const float* __restrict__ A, const float* __restrict__ B,
    float* __restrict__ C, int M, int N, int K) {
    int wid  = (blockIdx.x * blockDim.x + threadIdx.x) >> 5;   // wave-uniform
    int lane = threadIdx.x & 31;
    int ntn  = N >> 6;                 // 64-column groups
    int tm   = wid / ntn;              // 32-row group
    int tn   = wid - tm * ntn;
    if (tm >= (M >> 5)) return;        // uniform exit, EXEC stays all-ones

    int l16  = lane & 15;
    int half = lane >> 4;              // 0 -> holds K0/K1, 1 -> holds K2/K3
    const float* Ar0 = A + (size_t)(tm * 32 + l16) * K;
    const float* Ar1 = Ar0 + (size_t)16 * K;
    int bcol = tn * 64 + l16;

    v8f acc00 = {}, acc01 = {}, acc02 = {}, acc03 = {};
    v8f acc10 = {}, acc11 = {}, acc12 = {}, acc13 = {};

    for (int k = 0; k < K; k += 4) {
        int ka = k + 2 * half;
        v2f a0, a1, b0, b1, b2, b3;
        a0.x = Ar0[ka]; a0.y = Ar0[ka + 1];
        a1.x = Ar1[ka]; a1.y = Ar1[ka + 1];
        const float* Bp0 = B + (size_t)ka * N + bcol;   // row ka
        const float* Bp1 = Bp0 + N;                     // row ka+1
        b0.x = Bp0[0];  b0.y = Bp1[0];
        b1.x = Bp0[16]; b1.y = Bp1[16];
        b2.x = Bp0[32]; b2.y = Bp1[32];
        b3.x = Bp0[48]; b3.y = Bp1[48];

        acc00 = __builtin_amdgcn_wmma_f32_16x16x4_f32(false, a0, false, b0, (short)0, acc00, false, false);
        acc01 = __builtin_amdgcn_wmma_f32_16x16x4_f32(false, a0, false, b1, (short)0, acc01, false, false);
        acc02 = __builtin_amdgcn_wmma_f32_16x16x4_f32(false, a0, false, b2, (short)0, acc02, false, false);
        acc03 = __builtin_amdgcn_wmma_f32_16x16x4_f32(false, a0, false, b3, (short)0, acc03, false, false);
        acc10 = __builtin_amdgcn_wmma_f32_16x16x4_f32(false, a1, false, b0, (short)0, acc10, false, false);
        acc11 = __builtin_amdgcn_wmma_f32_16x16x4_f32(false, a1, false, b1, (short)0, acc11, false, false);
        acc12 = __builtin_amdgcn_wmma_f32_16x16x4_f32(false, a1, false, b2, (short)0, acc12, false, false);
        acc13 = __builtin_amdgcn_wmma_f32_16x16x4_f32(false, a1, false, b3, (short)0, acc13, false, false);
    }

    // C/D layout: VGPR r -> row = tileRow + r + 8*half, col = tileCol + l16
    v8f* accs[2][4] = {{&acc00, &acc01, &acc02, &acc03},
                       {&acc10, &acc11, &acc12, &acc13}};
#pragma unroll
    for (int i = 0; i < 2; i++) {
#pragma unroll
        for (int j = 0; j < 4; j++) {
            float* Cp = C + (size_t)(tm * 32 + i * 16 + half * 8) * N + tn * 64 + j * 16 + l16;
            v8f a = *accs[i][j];
#pragma unroll
            for (int r = 0; r < 8; r++) Cp[(size_t)r * N] = a[r];
        }
    }
}

// ------------------------------ CSR build ----------------------------------
__global__ void zero_i32(int* p, int n) {
    int i = blockIdx.x * blockDim.x + threadIdx.x;
    if (i < n) p[i] = 0;
}
__global__ void copy_i32(const int* s, int* d, int n) {
    int i = blockIdx.x * blockDim.x + threadIdx.x;
    if (i < n) d[i] = s[i];
}
__global__ void count_edges(const int* __restrict__ ei, int ne, int* __restrict__ counts) {
    int e = blockIdx.x * blockDim.x + threadIdx.x;
    if (e >= ne) return;
    atomicAdd(&counts[ei[ne + e]], 1);   // row 1 = dst
}
// single-block exclusive scan over n counts -> offsets[n+1]
__global__ __launch_bounds__(1024) void scan_offsets(const int* __restrict__ counts,
                                                     int* __restrict__ offsets, int n) {
    __shared__ int part[1024];
    int t = threadIdx.x;
    int per = (n + 1023) / 1024;
    int beg = t * per;
    int end = min(beg + per, n);
    int s = 0;
    for (int i = beg; i < end; i++) s += counts[i];
    part[t] = s;
    __syncthreads();
    for (int off = 1; off < 1024; off <<= 1) {
        int v = (t >= off) ? part[t - off] : 0;
        __syncthreads();
        part[t] += v;
        __syncthreads();
    }
    int run = (t == 0) ? 0 : part[t - 1];
    for (int i = beg; i < end; i++) { offsets[i] = run; run += counts[i]; }
    if (t == 0) offsets[n] = part[1023];
}
__global__ void scatter_edges(const int* __restrict__ ei, int ne,
                              int* __restrict__ cursor, int* __restrict__ esrc) {
    int e = blockIdx.x * blockDim.x + threadIdx.x;
    if (e >= ne) return;
    int d = ei[ne + e];
    int pos = atomicAdd(&cursor[d], 1);
    esrc[pos] = ei[e];
}

// --------------------------- attention pieces ------------------------------
// a_src/a_dst dots (EDIM=128 fixed); also init nmax with the self-loop value.
__global__ void att_dots(const float* __restrict__ hpre,
                         const float* __restrict__ attS, const float* __restrict__ attD,
                         float* __restrict__ a_src, float* __restrict__ a_dst,
                         float* __restrict__ nmax, int n, int H) {
    int i = blockIdx.x * blockDim.x + threadIdx.x;   // over n*H ; row = i*128
    if (i >= n * H) return;
    int h = i % H;
    const float4* r4 = (const float4*)(hpre + (size_t)i * 128);
    const float4* s4 = (const float4*)(attS + (size_t)h * 128);
    const float4* d4 = (const float4*)(attD + (size_t)h * 128);
    float ss = 0.f, dd = 0.f;
#pragma unroll 8
    for (int j = 0; j < 32; j++) {
        float4 x = r4[j], a = s4[j], b = d4[j];
        ss += x.x * a.x + x.y * a.y + x.z * a.z + x.w * a.w;
        dd += x.x * b.x + x.y * b.y + x.z * b.z + x.w * b.w;
    }
    a_src[i] = ss;
    a_dst[i] = dd;
    nmax[i] = lrelu(ss + dd);            // self-loop seeds the segment max
}
__global__ void edge_max(const int* __restrict__ ei, int ne,
                         const float* __restrict__ a_src, const float* __restrict__ a_dst,
                         float* __restrict__ nmax, int H) {
    int e = blockIdx.x * blockDim.x + threadIdx.x;
    if (e >= ne) return;
    int s = ei[e], d = ei[ne + e];
    for (int h = 0; h < H; h++)
        atomicMaxF(&nmax[d * H + h], lrelu(a_src[s * H + h] + a_dst[d * H + h]));
}
__global__ void init_sum(const float* __restrict__ a_src, const float* __restrict__ a_dst,
                         const float* __restrict__ nmax, float* __restrict__ nsum, int nh) {
    int i = blockIdx.x * blockDim.x + threadIdx.x;
    if (i >= nh) return;
    nsum[i] = __expf(lrelu(a_src[i] + a_dst[i]) - nmax[i]);   // self-loop term
}
__global__ void edge_sum(const int* __restrict__ ei, int ne,
                         const float* __restrict__ a_src, const float* __restrict__ a_dst,
                         const float* __restrict__ nmax, float* __restrict__ nsum, int H) {
    int e = blockIdx.x * blockDim.x + threadIdx.x;
    if (e >= ne) return;
    int s = ei[e], d = ei[ne + e];
    for (int h = 0; h < H; h++) {
        float v = lrelu(a_src[s * H + h] + a_dst[d * H + h]);
        atomicAdd(&nsum[d * H + h], __expf(v - nmax[d * H + h]));
    }
}
// One wave per (node, head): acc[128] striped 4 floats/lane; atomic-free CSR walk.
__global__ __launch_bounds__(256) void aggregate(
    const float* __restrict__ hpre, const float* __restrict__ a_src,
    const float* __restrict__ a_dst, const float* __restrict__ nmax,
    const float* __restrict__ nsum, const int* __restrict__ offsets,
    const int* __restrict__ esrc, const float* __restrict__ bias,
    float* __restrict__ out, int n, int H) {
    int wid  = (blockIdx.x * blockDim.x + threadIdx.x) >> 5;
    int lane = threadIdx.x & 31;
    if (wid >= n * H) return;
    int node = wid / H, h = wid - node * H;

    float m    = nmax[wid];
    float inv  = 1.0f / nsum[wid];
    float adst = a_dst[wid];

    // self loop
    float alpha = __expf(lrelu(a_src[wid] + adst) - m) * inv;
    float4 x = ((const float4*)(hpre + (size_t)wid * 128))[lane];
    float4 acc;
    acc.x = x.x * alpha; acc.y = x.y * alpha; acc.z = x.z * alpha; acc.w = x.w * alpha;

    int beg = offsets[node], end = offsets[node + 1];
    for (int j = beg; j < end; j++) {
        int s = esrc[j];
        float al = __expf(lrelu(a_src[s * H + h] + adst) - m) * inv;
        float4 y = ((const float4*)(hpre + ((size_t)s * H + h) * 128))[lane];
        acc.x += y.x * al; acc.y += y.y * al; acc.z += y.z * al; acc.w += y.w * al;
    }
    float4 b = ((const float4*)(bias + (size_t)h * 128))[lane];
    acc.x = fmaxf(acc.x + b.x, 0.f); acc.y = fmaxf(acc.y + b.y, 0.f);
    acc.z = fmaxf(acc.z + b.z, 0.f); acc.w = fmaxf(acc.w + b.w, 0.f);
    ((float4*)(out + (size_t)wid * 128))[lane] = acc;
}

// ------------------------------ final concat -------------------------------
__global__ void concat_out(const float* __restrict__ h2, const float* __restrict__ vm,
                           float* __restrict__ out, int V, long long flatN, int dvm) {
    long long total  = (long long)V * (flatN + dvm);
    long long stride = (long long)gridDim.x * blockDim.x;
    for (long long i = (long long)blockIdx.x * blockDim.x + threadIdx.x; i < total; i += stride) {
        long long row = i / (flatN + dvm);
        long long col = i - row * (flatN + dvm);
        out[i] = (col < flatN) ? h2[col] : vm[row * dvm + (col - flatN)];
    }
}

// ------------------------------- launcher ----------------------------------
extern "C" void kernel_launch(void* const* d_in, const int* in_sizes, int n_in,
                              void* d_out, int out_size, void* d_ws, size_t ws_size,
                              hipStream_t stream) {
    const float* x      = (const float*)d_in[0];   // [N,128]
    const float* vm     = (const float*)d_in[1];   // [V,64]
    const int*   ei     = (const int*)  d_in[2];   // [2,NE]
    const float* W1     = (const float*)d_in[3];   // [128,512]
    const float* attS1  = (const float*)d_in[4];   // [4,128]
    const float* attD1  = (const float*)d_in[5];
    const float* b1     = (const float*)d_in[6];   // [512]
    const float* W2     = (const float*)d_in[7];   // [512,128]
    const float* attS2  = (const float*)d_in[8];   // [1,128]
    const float* attD2  = (const float*)d_in[9];
    const float* b2     = (const float*)d_in[10];  // [128]

    const int DIN = 128, H1 = 4, ED = 128, HF1 = H1 * ED, DVM = 64;
    const int N  = in_sizes[0] / DIN;
    const int V  = in_sizes[1] / DVM;
    const int NE = in_sizes[2] / 2;

    // workspace carve-out
    char* base = (char*)d_ws;
    size_t off = 0;
    auto carve = [&](size_t bytes) -> char* {
        off = (off + 255) & ~(size_t)255;
        char* p = base + off;
        off += bytes;
        return p;
    };
    float* h1pre = (float*)carve((size_t)N * HF1 * 4);
    float* h1    = (float*)carve((size_t)N * HF1 * 4);
    float* h2pre = (float*)carve((size_t)N * ED * 4);
    float* h2    = (float*)carve((size_t)N * ED * 4);
    float* aS1   = (float*)carve((size_t)N * H1 * 4);
    float* aD1   = (float*)carve((size_t)N * H1 * 4);
    float* mx1   = (float*)carve((size_t)N * H1 * 4);
    float* sm1   = (float*)carve((size_t)N * H1 * 4);
    float* aS2   = (float*)carve((size_t)N * 4);
    float* aD2   = (float*)carve((size_t)N * 4);
    float* mx2   = (float*)carve((size_t)N * 4);
    float* sm2   = (float*)carve((size_t)N * 4);
    int*   offs  = (int*)  carve((size_t)(N + 1) * 4);
    int*   curs  = (int*)  carve((size_t)N * 4);
    int*   esrc  = (int*)  carve((size_t)NE * 4);
    (void)ws_size;

    const int T = 256;
    auto blk = [](long long n, int t) { return (int)((n + t - 1) / t); };

    // ---- CSR by destination (shared by both layers) ----
    zero_i32<<<blk(N, T), T, 0, stream>>>(curs, N);
    count_edges<<<blk(NE, T), T, 0, stream>>>(ei, NE, curs);
    scan_offsets<<<1, 1024, 0, stream>>>(curs, offs, N);
    copy_i32<<<blk(N, T), T, 0, stream>>>(offs, curs, N);
    scatter_edges<<<blk(NE, T), T, 0, stream>>>(ei, NE, curs, esrc);

    // ---- layer 1 ----
    {
        long long waves = (long long)(N / 32) * (HF1 / 64);   // 32x64 tile per wave
        gemm_wmma_f32<<<blk(waves * 32, T), T, 0, stream>>>(x, W1, h1pre, N, HF1, DIN);
        att_dots<<<blk((long long)N * H1, T), T, 0, stream>>>(h1pre, attS1, attD1, aS1, aD1, mx1, N, H1);
        edge_max<<<blk(NE, T), T, 0, stream>>>(ei, NE, aS1, aD1, mx1, H1);
        init_sum<<<blk((long long)N * H1, T), T, 0, stream>>>(aS1, aD1, mx1, sm1, N * H1);
        edge_sum<<<blk(NE, T), T, 0, stream>>>(ei, NE, aS1, aD1, mx1, sm1, H1);
        aggregate<<<blk((long long)N * H1 * 32, T), T, 0, stream>>>(
            h1pre, aS1, aD1, mx1, sm1, offs, esrc, b1, h1, N, H1);
    }

    // ---- layer 2 ----
    {
        long long waves = (long long)(N / 32) * (ED / 64);
        gemm_wmma_f32<<<blk(waves * 32, T), T, 0, stream>>>(h1, W2, h2pre, N, ED, HF1);
        att_dots<<<blk((long long)N, T), T, 0, stream>>>(h2pre, attS2, attD2, aS2, aD2, mx2, N, 1);
        edge_max<<<blk(NE, T), T, 0, stream>>>(ei, NE, aS2, aD2, mx2, 1);
        init_sum<<<blk((long long)N, T), T, 0, stream>>>(aS2, aD2, mx2, sm2, N);
        edge_sum<<<blk(NE, T), T, 0, stream>>>(ei, NE, aS2, aD2, mx2, sm2, 1);
        aggregate<<<blk((long long)N * 32, T), T, 0, stream>>>(
            h2pre, aS2, aD2, mx2, sm2, offs, esrc, b2, h2, N, 1);
    }

    // ---- broadcast + concat vm features ----
    concat_out<<<20000, T, 0, stream>>>(h2, vm, (float*)d_out, V, (long long)N * ED, DVM);
    (void)n_in; (void)out_size;
}